// GCN_49065706389779
// MI455X (gfx1250) — compile-verified
//
#include <hip/hip_runtime.h>

typedef __attribute__((ext_vector_type(2))) float v2f;
typedef __attribute__((ext_vector_type(8))) float v8f;

#define NN 100000
#define NE 1280000

// ---------------- degree / normalization ----------------

__global__ __launch_bounds__(256) void k_zero(float* __restrict__ p, int n) {
  int t = blockIdx.x * blockDim.x + threadIdx.x;
  if (t < n) p[t] = 0.0f;
}

__global__ __launch_bounds__(256) void k_deg(const long long* __restrict__ ei,
                                             float* __restrict__ deg, int E) {
  int e = blockIdx.x * blockDim.x + threadIdx.x;
  if (e < E) {
    long long d = ei[E + e];
    atomicAdd(&deg[d], 1.0f);
  }
}

__global__ __launch_bounds__(256) void k_dinv(float* __restrict__ deg, int n) {
  int t = blockIdx.x * blockDim.x + threadIdx.x;
  if (t < n) deg[t] = rsqrtf(deg[t] + 1.0f);  // +1 = self loop
}

// ---------------- f32 WMMA GEMM: C[M,N] = A[M,K] * B[K,N] ----------------
// One wave computes one 16x16 tile of C with V_WMMA_F32_16X16X4_F32.
// M%16==0, N%16==0, K%4==0 guaranteed by caller.
__global__ __launch_bounds__(256) void k_gemm_wmma_f32(
    const float* __restrict__ A, const float* __restrict__ B,
    float* __restrict__ C, int M, int K, int N) {
  int wave = (blockIdx.x * blockDim.x + threadIdx.x) >> 5;
  int lane = threadIdx.x & 31;
  int nTiles = N >> 4;
  int mTile = wave / nTiles;
  int nTile = wave - mTile * nTiles;
  if (mTile >= (M >> 4)) return;  // wave-uniform: EXEC stays all-1s for WMMA

  int half = lane >> 4;   // 0: K pair {0,1}, 1: K pair {2,3}
  int lid  = lane & 15;
  const float* Arow = A + (size_t)(mTile * 16 + lid) * K;
  const float* Bcol = B + (nTile * 16 + lid);

  v8f c = {};
  for (int k = 0; k < K; k += 4) {
    int kk = k + 2 * half;
    v2f a, b;
    a.x = Arow[kk];
    a.y = Arow[kk + 1];
    b.x = Bcol[(size_t)kk * N];
    b.y = Bcol[(size_t)(kk + 1) * N];
    c = __builtin_amdgcn_wmma_f32_16x16x4_f32(
        false, a, false, b, (short)0, c, false, false);
  }

  float* Crow = C + (size_t)(mTile * 16 + half * 8) * N + nTile * 16 + lid;
#pragma unroll
  for (int j = 0; j < 8; ++j) Crow[(size_t)j * N] = c[j];
}

// ------- self-loop init + bias fold: acc = dinv^2 * xw + b  (float4) -------

template <int C>
__global__ __launch_bounds__(256) void k_self_bias(const float* __restrict__ xw,
                                                   const float* __restrict__ dinv,
                                                   const float* __restrict__ b,
                                                   float* __restrict__ acc, int n) {
  int t = blockIdx.x * blockDim.x + threadIdx.x;  // one float4 per thread
  if (t >= n * (C / 4)) return;
  int node = t / (C / 4);
  int c4   = t - node * (C / 4);
  float di = dinv[node];
  float nrm = di * di;
  float4 v = reinterpret_cast<const float4*>(xw)[t];
  float4 bb = reinterpret_cast<const float4*>(b)[c4];
  float4 o;
  o.x = nrm * v.x + bb.x;
  o.y = nrm * v.y + bb.y;
  o.z = nrm * v.z + bb.z;
  o.w = nrm * v.w + bb.w;
  reinterpret_cast<float4*>(acc)[t] = o;
}

// ---------------- edge scatter: acc[dst] += dinv[s]*dinv[d]*xw[src] ----------
// Wave per edge; edge id forced into an SGPR via readfirstlane so the
// index/dinv loads become scalar (SMEM) and address math stays in SALU.

__global__ __launch_bounds__(256) void k_msg128(const long long* __restrict__ ei,
                                                const float* __restrict__ dinv,
                                                const float* __restrict__ xw,
                                                float* __restrict__ acc, int E) {
  int e = __builtin_amdgcn_readfirstlane(blockIdx.x * 8 + (threadIdx.x >> 5));
  int lane = threadIdx.x & 31;        // 4 channels per lane
  if (e >= E) return;
  long long s = ei[e], d = ei[E + e];
  float norm = dinv[s] * dinv[d];
  const float4 v = *reinterpret_cast<const float4*>(xw + (size_t)s * 128 + lane * 4);
  float* dst = acc + (size_t)d * 128 + lane * 4;
  atomicAdd(dst + 0, norm * v.x);
  atomicAdd(dst + 1, norm * v.y);
  atomicAdd(dst + 2, norm * v.z);
  atomicAdd(dst + 3, norm * v.w);
}

__global__ __launch_bounds__(256) void k_msg64(const long long* __restrict__ ei,
                                               const float* __restrict__ dinv,
                                               const float* __restrict__ xw,
                                               float* __restrict__ acc, int E) {
  int e = __builtin_amdgcn_readfirstlane(blockIdx.x * 8 + (threadIdx.x >> 5));
  int lane = threadIdx.x & 31;        // 2 channels per lane
  if (e >= E) return;
  long long s = ei[e], d = ei[E + e];
  float norm = dinv[s] * dinv[d];
  const float2 v = *reinterpret_cast<const float2*>(xw + (size_t)s * 64 + lane * 2);
  float* dst = acc + (size_t)d * 64 + lane * 2;
  atomicAdd(dst + 0, norm * v.x);
  atomicAdd(dst + 1, norm * v.y);
}

// ---------------- relu pass (layer 1 only, float4) ----------------

__global__ __launch_bounds__(256) void k_relu(float* __restrict__ acc, int n4) {
  int t = blockIdx.x * blockDim.x + threadIdx.x;
  if (t >= n4) return;
  float4 v = reinterpret_cast<float4*>(acc)[t];
  v.x = fmaxf(v.x, 0.0f);
  v.y = fmaxf(v.y, 0.0f);
  v.z = fmaxf(v.z, 0.0f);
  v.w = fmaxf(v.w, 0.0f);
  reinterpret_cast<float4*>(acc)[t] = v;
}

// ---------------- decode: scores[e] = dot(z[src], z[dst]) ----------------

__global__ __launch_bounds__(256) void k_score(const long long* __restrict__ ei,
                                               const float* __restrict__ z,
                                               float* __restrict__ out, int E) {
  int e = __builtin_amdgcn_readfirstlane(blockIdx.x * 8 + (threadIdx.x >> 5));
  int lane = threadIdx.x & 31;        // 2 channels per lane
  if (e >= E) return;
  long long s = ei[e], d = ei[E + e];
  const float2 zs = *reinterpret_cast<const float2*>(z + (size_t)s * 64 + lane * 2);
  const float2 zd = *reinterpret_cast<const float2*>(z + (size_t)d * 64 + lane * 2);
  float p = zs.x * zd.x + zs.y * zd.y;
#pragma unroll
  for (int off = 16; off > 0; off >>= 1) p += __shfl_xor(p, off, 32);
  if (lane == 0) out[e] = p;
}

// ---------------- launcher ----------------

extern "C" void kernel_launch(void* const* d_in, const int* in_sizes, int n_in,
                              void* d_out, int out_size, void* d_ws, size_t ws_size,
                              hipStream_t stream) {
  const float*     x  = (const float*)d_in[0];
  const long long* ei = (const long long*)d_in[1];
  const float*     W1 = (const float*)d_in[2];
  const float*     b1 = (const float*)d_in[3];
  const float*     W2 = (const float*)d_in[4];
  const float*     b2 = (const float*)d_in[5];
  float* scores = (float*)d_out;

  float* ws   = (float*)d_ws;
  float* dinv = ws;                               // [NN]
  float* bufA = ws + NN;                          // [NN*128]
  float* bufB = bufA + (size_t)NN * 128;          // [NN*128]
  float* xw1  = bufA;                             // layer-1 x@W1
  float* acc1 = bufB;                             // layer-1 aggregate -> h (in place)
  float* h    = bufB;
  float* xw2  = bufA;                             // recycled: h@W2 (NN*64)
  float* acc2 = bufA + (size_t)NN * 64;           // recycled: layer-2 aggregate -> z

  const int B = 256;
  const int gN     = (NN + B - 1) / B;
  const int gE     = (NE + B - 1) / B;
  const int gEwave = (NE + 7) / 8;                // wave (32 lanes) per edge
  const int gN128v = ((size_t)NN * 32 + B - 1) / B;   // N*128/4 float4s
  const int gN64v  = ((size_t)NN * 16 + B - 1) / B;   // N*64/4  float4s

  // normalization
  k_zero<<<gN, B, 0, stream>>>(dinv, NN);
  k_deg<<<gE, B, 0, stream>>>(ei, dinv, NE);
  k_dinv<<<gN, B, 0, stream>>>(dinv, NN);

  // layer 1: xw1 = x @ W1, aggregate (+b1 folded into init), relu
  {
    int waves = (NN / 16) * (128 / 16);
    k_gemm_wmma_f32<<<(waves + 7) / 8, B, 0, stream>>>(x, W1, xw1, NN, 64, 128);
  }
  k_self_bias<128><<<gN128v, B, 0, stream>>>(xw1, dinv, b1, acc1, NN);
  k_msg128<<<gEwave, B, 0, stream>>>(ei, dinv, xw1, acc1, NE);
  k_relu<<<gN128v, B, 0, stream>>>(acc1, NN * 32);

  // layer 2: xw2 = h @ W2, aggregate (+b2 folded into init); z = acc2
  {
    int waves = (NN / 16) * (64 / 16);
    k_gemm_wmma_f32<<<(waves + 7) / 8, B, 0, stream>>>(h, W2, xw2, NN, 128, 64);
  }
  k_self_bias<64><<<gN64v, B, 0, stream>>>(xw2, dinv, b2, acc2, NN);
  k_msg64<<<gEwave, B, 0, stream>>>(ei, dinv, xw2, acc2, NE);

  // decode
  k_score<<<gEwave, B, 0, stream>>>(ei, acc2, scores, NE);
}